// Kipfblock_28527172780471
// MI455X (gfx1250) — compile-verified
//
#include <hip/hip_runtime.h>
#include <math.h>

#define N_NODES 100000
#define N_EDGES 1600000
#define F 64
#define K_ORDER 8
#define LN_EPS 1e-5f

typedef __attribute__((ext_vector_type(2))) float v2f;
typedef __attribute__((ext_vector_type(8))) float v8f;

// ---------------------------------------------------------------- utilities
__global__ void zero_f32(float* __restrict__ p, int n) {
  int i = blockIdx.x * blockDim.x + threadIdx.x;
  if (i < n) p[i] = 0.0f;
}

// deg[r] += 1 for every non-self-loop edge (exact in fp32: deg <= 1.6M < 2^24)
__global__ void degree_kernel(const int* __restrict__ row,
                              const int* __restrict__ col,
                              float* __restrict__ deg) {
  int e = blockIdx.x * blockDim.x + threadIdx.x;
  if (e >= N_EDGES) return;
  int r = row[e], c = col[e];
  if (r != c) unsafeAtomicAdd(&deg[r], 1.0f);
}

__global__ void dinv_kernel(const float* __restrict__ deg,
                            float* __restrict__ dinv) {
  int i = blockIdx.x * blockDim.x + threadIdx.x;
  if (i >= N_NODES) return;
  float d = deg[i];
  dinv[i] = (d > 0.0f) ? rsqrtf(fmaxf(d, 1.0f)) : 0.0f;
}

// ------------------------------------------------------------------- SpMM
// out[row] += (-dinv[row]*dinv[col]) * h[col]; one wave32 per edge,
// each lane owns a float2 (32 lanes * 2 = 64 features).
__global__ void spmm_kernel(const int* __restrict__ row,
                            const int* __restrict__ col,
                            const float* __restrict__ dinv,
                            const float* __restrict__ h,
                            float* __restrict__ out) {
  int wid  = (blockIdx.x * blockDim.x + threadIdx.x) >> 5;
  int lane = threadIdx.x & 31;
  if (wid >= N_EDGES) return;
  int r = row[wid], c = col[wid];
  if (r == c) return;
  float w = -dinv[r] * dinv[c];
  v2f hv = *(const v2f*)(h + (size_t)c * F + lane * 2);
  float* o = out + (size_t)r * F + lane * 2;
  unsafeAtomicAdd(o,     w * hv.x);   // -> global_atomic_add_f32 (no return)
  unsafeAtomicAdd(o + 1, w * hv.y);
}

// Chebyshev recurrence elementwise step: p = 2*p - tkm2
__global__ void combine_kernel(float* __restrict__ p,
                               const float* __restrict__ tkm2, int n) {
  int i = blockIdx.x * blockDim.x + threadIdx.x;
  if (i < n) p[i] = 2.0f * p[i] - tkm2[i];
}

// --------------------------------------------------- WMMA fp32 GEMM (accum)
// acc[16x64 tile] = beta*acc + T_tile[16x64] @ Wk[64x64]
// One wave32 per 16-node tile. V_WMMA_F32_16X16X4_F32, K-loop of 16 steps.
// A 16x4 layout (ISA 7.12.2): lane&15 = M; half-wave selects K {0,1} vs {2,3}
//   -> contiguous float2 load from T row.
// B 4x16: row striped across lanes within a VGPR (transposed analogue).
// C/D 16x16: VGPR r holds (M = r + 8*(lane>>4), N = lane&15).
__global__ void wmma_gemm_kernel(const float* __restrict__ T,
                                 const float* __restrict__ Wk,
                                 float* __restrict__ acc, int beta) {
  int wavesPerBlock = blockDim.x >> 5;
  int tile = blockIdx.x * wavesPerBlock + (threadIdx.x >> 5);
  const int numTiles = N_NODES / 16;          // 6250 exactly
  if (tile >= numTiles) return;               // wave-uniform exit (EXEC stays all-1 for WMMA)
  int lane = threadIdx.x & 31;
  int m    = lane & 15;
  int half = lane >> 4;
  int n    = lane & 15;

  v8f c0 = {}, c1 = {}, c2 = {}, c3 = {};
  const float* trow = T + ((size_t)tile * 16 + m) * F;
  float* accbase = acc + (size_t)tile * 16 * F;

  if (beta) {
    #pragma unroll
    for (int r = 0; r < 8; ++r) {
      int mm = r + 8 * half;
      c0[r] = accbase[mm * F +  0 + n];
      c1[r] = accbase[mm * F + 16 + n];
      c2[r] = accbase[mm * F + 32 + n];
      c3[r] = accbase[mm * F + 48 + n];
    }
  }

  #pragma unroll
  for (int kk = 0; kk < F; kk += 4) {
    int k0 = kk + half * 2;
    v2f a = *(const v2f*)(trow + k0);
    const float* wr0 = Wk + (size_t)k0 * F;        // W row k0
    const float* wr1 = wr0 + F;                    // W row k0+1
    v2f b0 = { wr0[ 0 + n], wr1[ 0 + n] };
    v2f b1 = { wr0[16 + n], wr1[16 + n] };
    v2f b2 = { wr0[32 + n], wr1[32 + n] };
    v2f b3 = { wr0[48 + n], wr1[48 + n] };
    c0 = __builtin_amdgcn_wmma_f32_16x16x4_f32(false, a, false, b0, (short)0, c0, false, false);
    c1 = __builtin_amdgcn_wmma_f32_16x16x4_f32(false, a, false, b1, (short)0, c1, false, false);
    c2 = __builtin_amdgcn_wmma_f32_16x16x4_f32(false, a, false, b2, (short)0, c2, false, false);
    c3 = __builtin_amdgcn_wmma_f32_16x16x4_f32(false, a, false, b3, (short)0, c3, false, false);
  }

  #pragma unroll
  for (int r = 0; r < 8; ++r) {
    int mm = r + 8 * half;
    accbase[mm * F +  0 + n] = c0[r];
    accbase[mm * F + 16 + n] = c1[r];
    accbase[mm * F + 32 + n] = c2[r];
    accbase[mm * F + 48 + n] = c3[r];
  }
}

// ----------------------------------------------------- LayerNorm + ELU
// One wave32 per node; lane owns float2; 5-step shfl_xor tree (wave32).
__global__ void ln_elu_kernel(const float* __restrict__ acc,
                              const float* __restrict__ bias,
                              const float* __restrict__ lnw,
                              const float* __restrict__ lnb,
                              float* __restrict__ out) {
  int node = blockIdx.x * (blockDim.x >> 5) + (threadIdx.x >> 5);
  if (node >= N_NODES) return;
  int lane = threadIdx.x & 31;
  int f = lane * 2;
  v2f v = *(const v2f*)(acc + (size_t)node * F + f);
  v.x += bias[f];
  v.y += bias[f + 1];
  float s = v.x + v.y;
  #pragma unroll
  for (int off = 16; off > 0; off >>= 1) s += __shfl_xor(s, off, 32);
  float mean = s * (1.0f / F);
  float dx = v.x - mean, dy = v.y - mean;
  float q = dx * dx + dy * dy;
  #pragma unroll
  for (int off = 16; off > 0; off >>= 1) q += __shfl_xor(q, off, 32);
  float inv = rsqrtf(q * (1.0f / F) + LN_EPS);
  float hx = dx * inv * lnw[f]     + lnb[f];
  float hy = dy * inv * lnw[f + 1] + lnb[f + 1];
  hx = (hx > 0.0f) ? hx : expm1f(hx);   // ELU alpha=1
  hy = (hy > 0.0f) ? hy : expm1f(hy);
  v2f o = { hx, hy };
  *(v2f*)(out + (size_t)node * F + f) = o;
}

// ------------------------------------------------------------------ launch
extern "C" void kernel_launch(void* const* d_in, const int* in_sizes, int n_in,
                              void* d_out, int out_size, void* d_ws, size_t ws_size,
                              hipStream_t stream) {
  (void)in_sizes; (void)n_in; (void)out_size; (void)ws_size;

  const float* x      = (const float*)d_in[0];   // [N, 64]
  const int*   edge   = (const int*)d_in[1];     // [2, E]
  const float* weight = (const float*)d_in[2];   // [K, 64, 64]
  const float* bias   = (const float*)d_in[3];   // [64]
  const float* lnw    = (const float*)d_in[4];   // [64]
  const float* lnb    = (const float*)d_in[5];   // [64]
  const int* row = edge;
  const int* col = edge + N_EDGES;

  // Workspace: deg | dinv | b0 | b1 | b2   (acc lives in d_out)
  float* ws   = (float*)d_ws;
  float* deg  = ws;
  float* dinv = deg + N_NODES;
  float* b0   = dinv + N_NODES;
  float* b1   = b0 + (size_t)N_NODES * F;
  float* b2   = b1 + (size_t)N_NODES * F;
  float* acc  = (float*)d_out;                   // fully overwritten (beta=0 first)

  const int nfeat = N_NODES * F;
  dim3 blk(256);

  // --- normalization ---
  zero_f32<<<(N_NODES + 255) / 256, blk, 0, stream>>>(deg, N_NODES);
  degree_kernel<<<(N_EDGES + 255) / 256, blk, 0, stream>>>(row, col, deg);
  dinv_kernel<<<(N_NODES + 255) / 256, blk, 0, stream>>>(deg, dinv);

  // --- T1 = prop(x) ---
  int spmmGrid = (N_EDGES * 32 + 255) / 256;
  zero_f32<<<(nfeat + 255) / 256, blk, 0, stream>>>(b0, nfeat);
  spmm_kernel<<<spmmGrid, blk, 0, stream>>>(row, col, dinv, x, b0);

  // --- acc = T0@W0 + T1@W1 ---
  const int tiles = N_NODES / 16;
  dim3 gblk(128);                                // 4 waves/block
  int ggrid = (tiles + 3) / 4;
  wmma_gemm_kernel<<<ggrid, gblk, 0, stream>>>(x,  weight + 0 * F * F, acc, 0);
  wmma_gemm_kernel<<<ggrid, gblk, 0, stream>>>(b0, weight + 1 * F * F, acc, 1);

  // --- Chebyshev recurrence k = 2..7 ---
  const float* tkm2 = x;
  const float* tkm1 = b0;
  float* bufs[3] = { b0, b1, b2 };
  int next = 1;
  for (int k = 2; k < K_ORDER; ++k) {
    float* p = bufs[next];
    zero_f32<<<(nfeat + 255) / 256, blk, 0, stream>>>(p, nfeat);
    spmm_kernel<<<spmmGrid, blk, 0, stream>>>(row, col, dinv, tkm1, p);
    combine_kernel<<<(nfeat + 255) / 256, blk, 0, stream>>>(p, tkm2, nfeat);
    wmma_gemm_kernel<<<ggrid, gblk, 0, stream>>>(p, weight + (size_t)k * F * F, acc, 1);
    tkm2 = tkm1;
    tkm1 = p;
    next = (next + 1) % 3;
  }

  // --- LayerNorm + ELU (in-place safe: wave reads its node row before writing) ---
  ln_elu_kernel<<<(N_NODES + 7) / 8, blk, 0, stream>>>(acc, bias, lnw, lnb, (float*)d_out);
}